// DiffKS_37220186587561
// MI455X (gfx1250) — compile-verified
//
#include <hip/hip_runtime.h>
#include <hip/hip_bf16.h>
#include <math.h>

typedef __attribute__((ext_vector_type(2))) float v2f;
typedef __attribute__((ext_vector_type(8))) float v8f;

#define T_SAMPLES 44100
#define N_FRAMES  100
#define NB        6
#define NACT      7
#define EXC_ORDER 5
#define BURST_LEN 2048
#define NSOLVE    98          // N_FRAMES - 2 (tridiagonal system size)
#define NPAD      112         // 98 padded to multiple of 16
#define NCH       12          // 1 delay + 6 loop coeffs + 5 exc coeffs
#define CPAD      16          // channels padded to 16

// workspace layout (float offsets)
#define OFF_Y     0                         // [100][12]         = 1200
#define OFF_RHS   1216                      // [112][16] padded  = 1792
#define OFF_TINV  3072                      // [112][112] padded = 12544
#define OFF_M     15616                     // [112][16] padded  = 1792
#define OFF_V     17472                     // [44100][8]        = 352800
#define OFF_ZL    370272                    // 44100 ints
#define OFF_EXCA  414400                    // [2048][8]         = 16384
// total ~430784 floats = ~1.73 MB of ws

// ---------------------------------------------------------------------------
// Kernel A: constrained coefficients + spline RHS assembly
// ---------------------------------------------------------------------------
__global__ void ka_prep(const float* __restrict__ delay_f,
                        const float* __restrict__ raw_gain,
                        const float* __restrict__ raw_coeff,
                        const float* __restrict__ exc_c,
                        float* __restrict__ ws)
{
    int tid = threadIdx.x;
    float* Y   = ws + OFF_Y;     // [100][12] knot values
    float* RHS = ws + OFF_RHS;   // [112][16] padded rhs
    if (tid < N_FRAMES) {
        float gain = 1.0f / (1.0f + expf(-raw_gain[0]));
        float sb[NB]; float s = 0.f;
        for (int k = 0; k < NB; ++k) {
            sb[k] = 1.0f / (1.0f + expf(-raw_coeff[tid * NB + k]));
            s += sb[k];
        }
        Y[tid * NCH + 0] = delay_f[tid];
        for (int k = 0; k < NB; ++k)        Y[tid * NCH + 1 + k] = sb[k] / s * gain;
        for (int k = 0; k < EXC_ORDER; ++k) Y[tid * NCH + 7 + k] = exc_c[tid * EXC_ORDER + k];
    }
    __syncthreads();
    for (int i = tid; i < NPAD * CPAD; i += blockDim.x) RHS[i] = 0.f;
    __syncthreads();
    const float h = 1.0f / 99.0f;
    const float s6h2 = 6.0f / (h * h);
    if (tid < NSOLVE) {
        for (int c = 0; c < NCH; ++c) {
            float r = (Y[tid * NCH + c] - 2.f * Y[(tid + 1) * NCH + c]
                       + Y[(tid + 2) * NCH + c]) * s6h2;
            RHS[tid * CPAD + c] = r;
        }
    }
}

// ---------------------------------------------------------------------------
// Kernel B: inverse of tridiag(1,4,1)_98 via 98 parallel Thomas solves
// ---------------------------------------------------------------------------
__global__ void kb_tinv(float* __restrict__ ws)
{
    __shared__ float cp[NSOLVE];
    int tid = threadIdx.x;
    float* Tinv = ws + OFF_TINV;   // [112][112], row-major, zero-padded
    for (int i = tid; i < NPAD * NPAD; i += blockDim.x) Tinv[i] = 0.f;
    if (tid == 0) {
        cp[0] = 0.25f;
        for (int i = 1; i < NSOLVE; ++i) cp[i] = 1.0f / (4.0f - cp[i - 1]);
    }
    __syncthreads();
    if (tid < NSOLVE) {
        float x[NSOLVE];
        // forward sweep on e_tid
        x[0] = (tid == 0 ? 1.0f : 0.0f) * 0.25f;
        for (int i = 1; i < NSOLVE; ++i) {
            float r = (i == tid) ? 1.0f : 0.0f;
            x[i] = (r - x[i - 1]) * cp[i];
        }
        // back substitution
        for (int i = NSOLVE - 2; i >= 0; --i) x[i] = x[i] - cp[i] * x[i + 1];
        for (int i = 0; i < NSOLVE; ++i) Tinv[i * NPAD + tid] = x[i];
    }
}

// ---------------------------------------------------------------------------
// Kernel C: M_in = Tinv @ RHS  via V_WMMA_F32_16X16X4_F32 (7 waves, 28 K-steps)
// A 16x4 layout: lanes 0-15 hold K0/K1, lanes 16-31 hold K2/K3 (ISA 7.12.2)
// B 4x16 / D 16x16: rows striped across lanes, halves split at lane 16
// ---------------------------------------------------------------------------
__global__ void kc_gemm(float* __restrict__ ws)
{
    const float* A = ws + OFF_TINV;  // [112][112]
    const float* B = ws + OFF_RHS;   // [112][16]
    float*       D = ws + OFF_M;     // [112][16]
    int lane = threadIdx.x & 31;
    int w    = threadIdx.x >> 5;     // tile row 0..6
    int half = lane >> 4;            // 0: lanes 0-15, 1: lanes 16-31
    int l    = lane & 15;
    v8f acc = {};
    for (int kk = 0; kk < NPAD / 4; ++kk) {
        int k0 = kk * 4 + 2 * half;
        v2f a, b;
        a.x = A[(w * 16 + l) * NPAD + k0];
        a.y = A[(w * 16 + l) * NPAD + k0 + 1];
        b.x = B[(k0    ) * CPAD + l];
        b.y = B[(k0 + 1) * CPAD + l];
        acc = __builtin_amdgcn_wmma_f32_16x16x4_f32(
            /*neg_a=*/false, a, /*neg_b=*/false, b,
            /*c_mod=*/(short)0, acc, /*reuse_a=*/false, /*reuse_b=*/false);
    }
#pragma unroll
    for (int r = 0; r < 8; ++r)
        D[(w * 16 + r + 8 * half) * CPAD + l] = acc[r];
}

// ---------------------------------------------------------------------------
// Kernel D: spline evaluation at 44100 points + phase-delay compensation -> V, z_l
// ---------------------------------------------------------------------------
__global__ void kd_params(float* __restrict__ ws)
{
    int t = blockIdx.x * blockDim.x + threadIdx.x;
    if (t >= T_SAMPLES) return;
    const float* Y = ws + OFF_Y;    // knot values [100][12]
    const float* M = ws + OFF_M;    // interior 2nd derivs [98][16] (row i = knot i+1)
    float* V    = ws + OFF_V;
    int*   zl   = (int*)(ws + OFF_ZL);
    float* excA = ws + OFF_EXCA;

    const float h = 1.0f / 99.0f;
    float tt = (float)t / (float)(T_SAMPLES - 1);
    int idx = (int)floorf(tt * 99.0f);
    if (idx < 0) idx = 0;
    if (idx > 98) idx = 98;
    float dt = tt - (float)idx * h;

    float ch[NCH];
    for (int c = 0; c < NCH; ++c) {
        float y0 = Y[idx * NCH + c];
        float y1 = Y[(idx + 1) * NCH + c];
        float M0 = (idx == 0)      ? 0.f : M[(idx - 1) * CPAD + c];
        float M1 = (idx + 1 == 99) ? 0.f : M[idx * CPAD + c];
        float bb = (y1 - y0) / h - h * (2.f * M0 + M1) * (1.f / 6.f);
        float cc = 0.5f * M0;
        float dd = (M1 - M0) / (6.f * h);
        ch[c] = y0 + dt * (bb + dt * (cc + dt * dd));
    }

    float delay = ch[0];
    float b[NB];
    for (int k = 0; k < NB; ++k) b[k] = ch[1 + k];

    float omega = 2.0f * (float)M_PI / delay;
    float sw, cw;
    __sincosf(omega, &sw, &cw);
    // Chebyshev recurrence for cos(k*w), sin(k*w)
    float ck_m1 = 1.f, sk_m1 = 0.f;
    float ck = cw, sk = sw;
    float re = b[0];
    float im = 0.f;
    re += b[1] * ck; im -= b[1] * sk;
    for (int k = 2; k < NB; ++k) {
        float cn = 2.f * cw * ck - ck_m1;
        float sn = 2.f * cw * sk - sk_m1;
        ck_m1 = ck; sk_m1 = sk; ck = cn; sk = sn;
        re += b[k] * ck; im -= b[k] * sk;
    }
    float p_a = -atan2f(im, re) / omega;
    float d   = delay - (1.0f + p_a);
    float fd  = floorf(d);
    float alfa = d - fd;
    zl[t] = (int)fd;

    float v[8];
    v[0] = -(1.f - alfa) * b[0];
    for (int j = 1; j < NB; ++j) v[j] = -(alfa * b[j - 1] + (1.f - alfa) * b[j]);
    v[6] = -alfa * b[NB - 1];
    v[7] = 0.f;
    float4* Vp = (float4*)(V + t * 8);
    Vp[0] = make_float4(v[0], v[1], v[2], v[3]);
    Vp[1] = make_float4(v[4], v[5], v[6], v[7]);

    if (t < BURST_LEN) {
        for (int k = 0; k < EXC_ORDER; ++k) excA[t * 8 + k] = ch[7 + k];
        for (int k = EXC_ORDER; k < 8; ++k) excA[t * 8 + k] = 0.f;
    }
}

// ---------------------------------------------------------------------------
// Kernel F: excitation LPC (lane 0, serial) + Karplus-Strong loop.
// One wave32; full 44100-sample history lives in LDS (176KB < 320KB/WGP).
// Min feedback lag = z_l+1 >= ~94 > 32  =>  32-sample chunks are independent.
// ---------------------------------------------------------------------------
__global__ void kf_loop(const float* __restrict__ excitation,
                        const float* __restrict__ ws_c,
                        float* __restrict__ out)
{
    extern __shared__ float lds[];
    float* ybuf = lds;                 // [44100]
    float* xbuf = lds + T_SAMPLES;     // [2048]
    const float* excA = ws_c + OFF_EXCA;
    const float* V    = ws_c + OFF_V;
    const int*   zl   = (const int*)(ws_c + OFF_ZL);
    int lane = threadIdx.x;

    if (lane == 0) {
        float b0 = 0.f, b1 = 0.f, b2 = 0.f, b3 = 0.f, b4 = 0.f;
        for (int t = 0; t < BURST_LEN; ++t) {
            const float* a = excA + t * 8;
            float y = excitation[t]
                    - (a[0] * b0 + a[1] * b1 + a[2] * b2 + a[3] * b3 + a[4] * b4);
            b4 = b3; b3 = b2; b2 = b1; b1 = b0; b0 = y;
            xbuf[t] = y;
        }
    }
    __syncthreads();

    for (int t0 = 0; t0 < T_SAMPLES; t0 += 32) {
        int t = t0 + lane;
        if (t < T_SAMPLES) {
            const float4* Vp = (const float4*)(V + t * 8);
            float4 v0 = Vp[0];
            float4 v1 = Vp[1];
            int base = t - 1 - zl[t];
            float vv[7] = { v0.x, v0.y, v0.z, v0.w, v1.x, v1.y, v1.z };
            float acc = 0.f;
#pragma unroll
            for (int j = 0; j < 7; ++j) {
                int ix = base - j;
                float yv = (ix >= 0) ? ybuf[ix] : 0.f;
                acc += vv[j] * yv;
            }
            float xv = (t < BURST_LEN) ? xbuf[t] : 0.f;
            float yt = xv - acc;
            ybuf[t] = yt;
            out[t] = yt;
        }
        __syncthreads();
    }
}

// ---------------------------------------------------------------------------
extern "C" void kernel_launch(void* const* d_in, const int* in_sizes, int n_in,
                              void* d_out, int out_size, void* d_ws, size_t ws_size,
                              hipStream_t stream)
{
    const float* delay_f    = (const float*)d_in[0];
    const float* raw_gain   = (const float*)d_in[1];
    const float* raw_coeff  = (const float*)d_in[2];
    const float* exc_c      = (const float*)d_in[3];
    const float* excitation = (const float*)d_in[4];
    float* ws  = (float*)d_ws;
    float* out = (float*)d_out;

    ka_prep<<<1, 128, 0, stream>>>(delay_f, raw_gain, raw_coeff, exc_c, ws);
    kb_tinv<<<1, 128, 0, stream>>>(ws);
    kc_gemm<<<1, 224, 0, stream>>>(ws);   // 7 waves, one 16-row tile each
    kd_params<<<(T_SAMPLES + 255) / 256, 256, 0, stream>>>(ws);

    size_t ldsbytes = (size_t)(T_SAMPLES + BURST_LEN) * sizeof(float); // ~184 KB
    kf_loop<<<1, 32, ldsbytes, stream>>>(excitation, ws, out);
}